// LinearAttention_29472065585179
// MI455X (gfx1250) — compile-verified
//
#include <hip/hip_runtime.h>

// LinearAttention (b=32, c=256, n=4096, heads=4, dim_head=32) for MI455X gfx1250.
// Memory-bound (~1 GB traffic vs 37 GFLOP) -> fp32 precision everywhere, with
// V_WMMA_F32_16X16X4_F32 for all GEMM stages (exact fp32 accumulate).
// The attention-core k/v operands are staged into LDS by the Tensor Data Mover
// (tensor_load_to_lds, double-buffered, TENSORcnt-tracked) to eliminate the
// 16KB-strided 4-byte gathers that the WMMA fragment layout would otherwise
// force on global memory. TDM LDS padding (3 DWORDs per 128-DWORD row) makes
// the strided LDS fragment reads bank-conflict-free (131 = 3 mod 64 banks).
//
// Workspace layout (floats):
//   qkv  : [32][384][4096]   (q rows 0..127, k rows 128..255, v rows 256..383)
//   attn : [32][128][4096]
//   stats: [32][2]           (sum, sumsq per batch; atomically accumulated)

typedef __attribute__((ext_vector_type(2))) float v2f;
typedef __attribute__((ext_vector_type(8))) float v8f;
typedef __attribute__((ext_vector_type(4))) unsigned int v4u;
typedef __attribute__((ext_vector_type(8))) int v8i;
typedef __attribute__((ext_vector_type(4))) int v4i;

#define B_     32
#define C_     256
#define N_     4096
#define HEADS_ 4
#define DH_    32
#define HID_   128
#define QKVO_  384
#define NCHUNK 128          // n-columns staged per TDM transfer
#define RS_    131          // LDS row stride: 128 + 3 pad DWORDs (conflict-free)

static __device__ __forceinline__ v8f wmma_k4(v2f a, v2f b, v8f c) {
  // D = A(16x4 f32) * B(4x16 f32) + C(16x16 f32)
  return __builtin_amdgcn_wmma_f32_16x16x4_f32(
      /*neg_a=*/false, a, /*neg_b=*/false, b,
      /*c_mod=*/(short)0, c, /*reuse_a=*/false, /*reuse_b=*/false);
}

static __device__ __forceinline__ float wave_max32(float v) {
  for (int m = 16; m >= 1; m >>= 1) v = fmaxf(v, __shfl_xor(v, m, 32));
  return v;
}
static __device__ __forceinline__ float wave_sum32(float v) {
  for (int m = 16; m >= 1; m >>= 1) v += __shfl_xor(v, m, 32);
  return v;
}

// ---------------------------------------------------------------------------
// TDM: DMA a 3-D tile {x:128 cols, y:32 rows, z:2 matrices (k then v)} from
// global (row stride 4096 floats, z stride 128*4096 floats) into LDS with
// 3-DWORD padding after every 128-DWORD row. D# built per ISA 8.3-8.6.
// This toolchain (clang-23/therock) exposes the 6-argument builtin:
//   (uint32x4 g0, int32x8 g1, int32x4 g2, int32x4 g3, int32x8 g4, i32 cpol)
static __device__ __forceinline__ void tdm_load_kv(const float* gptr,
                                                   unsigned int lds_byte) {
  const unsigned long long ga = (unsigned long long)(uintptr_t)gptr;
  v4u g0;
  g0[0] = 1u;                                            // count=1, user mode
  g0[1] = lds_byte;                                      // lds_addr
  g0[2] = (unsigned int)ga;                              // global_addr[31:0]
  g0[3] = (unsigned int)((ga >> 32) & 0x01FFFFFFu)       // global_addr[56:32]
          | (2u << 30);                                  // type=2 ("image")
  v8i g1;
  g1[0] = (int)((2u << 16)      // data_size = 4 bytes
              | (1u << 20)      // pad_enable
              | (6u << 22)      // pad_interval: 128 DWORDs
              | (2u << 25));    // pad_amount: 3 DWORDs
  g1[1] = (int)(4096u << 16);   // tensor_dim0 = 4096 (bits 79:48, low half)
  g1[2] = (int)(32u << 16);     // tensor_dim0 hi=0 | tensor_dim1 = 32
  g1[3] = (int)(128u << 16);    // tensor_dim1 hi=0 | tile_dim0 = 128
  g1[4] = (int)(32u | (2u << 16)); // tile_dim1 = 32 | tile_dim2 = 2
  g1[5] = 4096;                 // tensor_dim0_stride[31:0]
  g1[6] = 0;                    // d0_stride hi | d1_stride[15:0] (524288 lo=0)
  g1[7] = 8;                    // tensor_dim1_stride[47:16] = 524288 >> 16
  v4i g2;
  g2[0] = 2;                    // tensor_dim2 = 2
  g2[1] = 0; g2[2] = 0; g2[3] = 0;
  v4i g3 = {0, 0, 0, 0};
  v8i g4 = {0, 0, 0, 0, 0, 0, 0, 0};   // unused trailing group (zero-filled)
  __builtin_amdgcn_tensor_load_to_lds(g0, g1, g2, g3, g4, /*cpol=*/0);
}

// ---------------------------------------------------------------------------
__global__ void zero_stats_kernel(float* stats) {
  if (threadIdx.x < 2 * B_) stats[threadIdx.x] = 0.0f;
}

// ---------------------------------------------------------------------------
// QKV projection: qkv[b][o][n] = sum_c w_qkv[o][c] * x[b][c][n]
__global__ __launch_bounds__(256) void qkv_proj_kernel(
    const float* __restrict__ x, const float* __restrict__ w,
    float* __restrict__ qkv) {
  const int b    = blockIdx.x;
  const int o0   = blockIdx.y << 4;
  const int lane = threadIdx.x & 31;
  const int wv   = threadIdx.x >> 5;
  const int n0   = (blockIdx.z * 8 + wv) << 4;
  const int mrow = lane & 15;
  const int kk   = (lane >> 4) << 1;
  const int mhi  = (lane >> 4) << 3;

  const float* xb = x + (size_t)b * C_ * N_;
  const float* wr = w + (size_t)(o0 + mrow) * C_;

  v8f acc = {};
  for (int k = 0; k < C_; k += 4) {
    v2f a, bb;
    a.x  = wr[k + kk];
    a.y  = wr[k + kk + 1];
    bb.x = xb[(size_t)(k + kk) * N_ + n0 + mrow];
    bb.y = xb[(size_t)(k + kk + 1) * N_ + n0 + mrow];
    acc  = wmma_k4(a, bb, acc);
  }
  float* dst = qkv + ((size_t)b * QKVO_ + o0) * (size_t)N_ + n0;
  for (int r = 0; r < 8; ++r) dst[(size_t)(r + mhi) * N_ + mrow] = acc[r];
}

// ---------------------------------------------------------------------------
// Attention core: one workgroup (256 threads, 8 waves) per (b, h).
__global__ __launch_bounds__(256) void attn_core_kernel(
    const float* __restrict__ qkv, float* __restrict__ attn) {
  const int bh   = blockIdx.x;
  const int b    = bh >> 2;
  const int h    = bh & 3;
  const int lane = threadIdx.x & 31;
  const int wv   = threadIdx.x >> 5;
  const int mrow = lane & 15;
  const int kk   = (lane >> 4) << 1;
  const int mhi  = (lane >> 4) << 3;

  const float* qmat = qkv + ((size_t)b * QKVO_ + 0   + h * DH_) * (size_t)N_;
  const float* kmat = qkv + ((size_t)b * QKVO_ + 128 + h * DH_) * (size_t)N_;

  __shared__ float kv[2][2][DH_][RS_];   // [buf][k/v][row][col+pad]
  __shared__ float smax[DH_];
  __shared__ float ssum[DH_];
  __shared__ float ctx[DH_][DH_ + 1];

  // Kick off the first k/v tile DMA; it overlaps the softmax-stats pass.
  if (wv == 0)
    tdm_load_kv(kmat, (unsigned int)(uintptr_t)&kv[0][0][0][0]);

  // ---- Phase A: softmax stats over n for each k row -----------------------
  for (int d = wv; d < DH_; d += 8) {
    const float* row = kmat + (size_t)d * N_;
    float m = -3.0e38f;
    for (int n = lane; n < N_; n += 32) m = fmaxf(m, row[n]);
    m = wave_max32(m);
    float s = 0.0f;
    for (int n = lane; n < N_; n += 32) s += __expf(row[n] - m);
    s = wave_sum32(s);
    if (lane == 0) { smax[d] = m; ssum[d] = s; }
  }
  __syncthreads();

  // ---- Phase B: context[d][e] = sum_n softmax(k)[d][n] * v[e][n] ----------
  // TDM double-buffered pipeline: wave 0 issues chunk i+1, waits for chunk i
  // (TDM completes in order per wave), barrier, 8 waves consume LDS tile.
  {
    const int t     = wv >> 1;            // output tile 0..3 (2x2 of 16x16)
    const int d0    = (t >> 1) << 4;
    const int e0    = (t & 1) << 4;
    const int khalf = wv & 1;             // split chunk columns across pair
    const int da    = d0 + mrow;
    const float mx  = smax[da];

    v8f acc = {};
    for (int ch = 0; ch < N_ / NCHUNK; ++ch) {
      if (wv == 0) {
        if (ch + 1 < N_ / NCHUNK) {
          tdm_load_kv(kmat + (size_t)(ch + 1) * NCHUNK,
                      (unsigned int)(uintptr_t)&kv[(ch + 1) & 1][0][0][0]);
          __builtin_amdgcn_s_wait_tensorcnt(1);  // chunk ch has landed
        } else {
          __builtin_amdgcn_s_wait_tensorcnt(0);
        }
      }
      __syncthreads();
      const float(*kb)[RS_] = kv[ch & 1][0];
      const float(*vb)[RS_] = kv[ch & 1][1];
      const int nb = khalf << 6;
      for (int nn = nb; nn < nb + 64; nn += 4) {
        v2f a, bb;
        a.x  = __expf(kb[da][nn + kk] - mx);
        a.y  = __expf(kb[da][nn + kk + 1] - mx);
        bb.x = vb[e0 + mrow][nn + kk];
        bb.y = vb[e0 + mrow][nn + kk + 1];
        acc  = wmma_k4(a, bb, acc);
      }
      __syncthreads();
    }

    if (khalf == 1) {
      for (int r = 0; r < 8; ++r) ctx[d0 + r + mhi][e0 + mrow] = acc[r];
    }
    __syncthreads();
    if (khalf == 0) {
      for (int r = 0; r < 8; ++r) {
        const int d = d0 + r + mhi;
        ctx[d][e0 + mrow] = (acc[r] + ctx[d][e0 + mrow]) / ssum[d];
      }
    }
    __syncthreads();
  }

  // ---- Phase C: out[e][n] = sum_d ctx[d][e] * (softmax_d(q)*scale)[d][n] --
  const float scale = 0.17677669529663687f;  // 32^-0.5
  float* obase = attn + ((size_t)b * HID_ + h * DH_) * (size_t)N_;
  for (int tile = wv; tile < 2 * (N_ / 16); tile += 8) {
    const int e0  = (tile & 1) << 4;
    const int n0  = (tile >> 1) << 4;
    const int col = n0 + mrow;
    // per-column softmax stats over the 32 q rows (held in registers)
    float m = -3.0e38f;
    for (int d = 0; d < DH_; ++d) m = fmaxf(m, qmat[(size_t)d * N_ + col]);
    float s = 0.0f;
    for (int d = 0; d < DH_; ++d) s += __expf(qmat[(size_t)d * N_ + col] - m);
    const float inv = scale / s;

    v8f acc = {};
    for (int k = 0; k < DH_; k += 4) {
      v2f a, bb;
      a.x  = ctx[k + kk][e0 + mrow];      // A = context^T (e x d)
      a.y  = ctx[k + kk + 1][e0 + mrow];
      bb.x = __expf(qmat[(size_t)(k + kk) * N_ + col] - m) * inv;
      bb.y = __expf(qmat[(size_t)(k + kk + 1) * N_ + col] - m) * inv;
      acc  = wmma_k4(a, bb, acc);
    }
    for (int r = 0; r < 8; ++r)
      obase[(size_t)(e0 + r + mhi) * N_ + n0 + mrow] = acc[r];
  }
}

// ---------------------------------------------------------------------------
// Output projection + bias, plus per-batch sum/sumsq for the norm.
__global__ __launch_bounds__(256) void out_proj_kernel(
    const float* __restrict__ attn, const float* __restrict__ w,
    const float* __restrict__ bias, float* __restrict__ out,
    float* __restrict__ stats) {
  const int b    = blockIdx.x;
  const int o0   = blockIdx.y << 4;
  const int lane = threadIdx.x & 31;
  const int wv   = threadIdx.x >> 5;
  const int n0   = (blockIdx.z * 8 + wv) << 4;
  const int mrow = lane & 15;
  const int kk   = (lane >> 4) << 1;
  const int mhi  = (lane >> 4) << 3;

  const float* ab = attn + (size_t)b * HID_ * N_;
  const float* wr = w + (size_t)(o0 + mrow) * HID_;

  v8f acc = {};
  for (int k = 0; k < HID_; k += 4) {
    v2f a, bb;
    a.x  = wr[k + kk];
    a.y  = wr[k + kk + 1];
    bb.x = ab[(size_t)(k + kk) * N_ + n0 + mrow];
    bb.y = ab[(size_t)(k + kk + 1) * N_ + n0 + mrow];
    acc  = wmma_k4(a, bb, acc);
  }
  float* dst = out + ((size_t)b * C_ + o0) * (size_t)N_ + n0;
  float s1 = 0.0f, s2 = 0.0f;
  for (int r = 0; r < 8; ++r) {
    const float v = acc[r] + bias[o0 + r + mhi];
    dst[(size_t)(r + mhi) * N_ + mrow] = v;
    s1 += v;
    s2 += v * v;
  }
  s1 = wave_sum32(s1);
  s2 = wave_sum32(s2);
  if (lane == 0) {
    atomicAdd(&stats[b * 2 + 0], s1);
    atomicAdd(&stats[b * 2 + 1], s2);
  }
}

// ---------------------------------------------------------------------------
// Whole-sample normalization + affine (GroupNorm with 1 group).
__global__ __launch_bounds__(256) void group_norm_kernel(
    float* __restrict__ out, const float* __restrict__ stats,
    const float* __restrict__ gw, const float* __restrict__ gb) {
  const int b = blockIdx.x;
  const float invM = 1.0f / (float)(C_ * N_);
  const float mean = stats[b * 2 + 0] * invM;
  const float var  = stats[b * 2 + 1] * invM - mean * mean;
  const float inv  = rsqrtf(var + 1e-5f);
  float* base = out + (size_t)b * C_ * N_;
  const int stride = gridDim.y * blockDim.x;
  for (int i = blockIdx.y * blockDim.x + threadIdx.x; i < C_ * N_; i += stride) {
    const int o = i >> 12;  // i / N_
    base[i] = (base[i] - mean) * inv * gw[o] + gb[o];
  }
}

// ---------------------------------------------------------------------------
extern "C" void kernel_launch(void* const* d_in, const int* in_sizes, int n_in,
                              void* d_out, int out_size, void* d_ws,
                              size_t ws_size, hipStream_t stream) {
  (void)in_sizes; (void)n_in; (void)out_size; (void)ws_size;

  const float* x     = (const float*)d_in[0];
  const float* w_qkv = (const float*)d_in[1];
  const float* w_out = (const float*)d_in[2];
  const float* b_out = (const float*)d_in[3];
  const float* gn_w  = (const float*)d_in[4];
  const float* gn_b  = (const float*)d_in[5];
  float* out = (float*)d_out;

  float* qkv   = (float*)d_ws;                                  // 32*384*4096
  float* attn  = qkv + (size_t)B_ * QKVO_ * N_;                 // 32*128*4096
  float* stats = attn + (size_t)B_ * HID_ * N_;                 // 32*2

  zero_stats_kernel<<<1, 64, 0, stream>>>(stats);
  qkv_proj_kernel<<<dim3(B_, QKVO_ / 16, N_ / 128), 256, 0, stream>>>(
      x, w_qkv, qkv);
  attn_core_kernel<<<dim3(B_ * HEADS_), 256, 0, stream>>>(qkv, attn);
  out_proj_kernel<<<dim3(B_, C_ / 16, N_ / 128), 256, 0, stream>>>(
      attn, w_out, b_out, out, stats);
  group_norm_kernel<<<dim3(B_, 64), 256, 0, stream>>>(out, stats, gn_w, gn_b);
}